// KnowledgeGuidedTransform_49065706390179
// MI455X (gfx1250) — compile-verified
//
#include <hip/hip_runtime.h>
#include <hip/hip_bf16.h>

// ---------------------------------------------------------------------------
// KnowledgeGuidedTransform — MI455X (gfx1250, wave32) implementation
//
// Algebraic restructuring: scatter-mean over edges is linear, so with
//   S[o,l]   = #edges mapping lab l -> organ o            (16 x 64)
//   K[o,f]   = sum_{l} S[o,l]*(h_conc[l,f]+b_lab[f]) + sum_{e in o} relmsg[e,f]
//   M2T[f',g]= sum_f W_org[g,f] * W_lab[f,f']             (256 x 256)
//   Kp[o,g]  = sum_f W_org[g,f] * K[o,f]                  (16 x 256)
// we get
//   organ_states[b,t,o,g] =
//       mask[b,t]/max(cnt[o],1) * ( (M2T^T @ (S @ X[b,t]))[o,g] + Kp[o,g] )
//       + W_org_b[g]
// Only large read: lab_features (67 MB) -> HBM-bound. f32 WMMA keeps f32 math.
// ---------------------------------------------------------------------------

typedef float v2f __attribute__((ext_vector_type(2)));
typedef float v8f __attribute__((ext_vector_type(8)));

__device__ __forceinline__ v8f wmma_f32(v2f a, v2f b, v8f c) {
  return __builtin_amdgcn_wmma_f32_16x16x4_f32(false, a, false, b, (short)0, c,
                                               false, false);
}

// problem dims (fixed by the reference)
constexpr int BT   = 16 * 64;   // B*T
constexpr int LDIM = 64;
constexpr int FDIM = 256;
constexpr int ODIM = 16;
constexpr int EDIM = 256;
constexpr int CDIM = 768;
constexpr int GF   = 256;       // organ_feat

// workspace layout (float offsets)
constexpr int OFF_S      = 0;                         // 16*64   = 1024
constexpr int OFF_CNT    = OFF_S + ODIM * LDIM;       // 16
constexpr int OFF_RELAGG = OFF_CNT + ODIM;            // 16*768  = 12288
constexpr int OFF_CBIAS  = OFF_RELAGG + ODIM * CDIM;  // 64*256  = 16384
constexpr int OFF_KMAT   = OFF_CBIAS + LDIM * FDIM;   // 16*256  = 4096
constexpr int OFF_KP     = OFF_KMAT + ODIM * FDIM;    // 16*256  = 4096
constexpr int OFF_M2T    = OFF_KP + ODIM * GF;        // 256*256 = 65536

// ---------------------------------------------------------------------------
__global__ __launch_bounds__(256) void kgt_zero(float* __restrict__ p, int n) {
  int i = blockIdx.x * 256 + threadIdx.x;
  if (i < n) p[i] = 0.0f;
}

// one block of 768 threads: builds S (count matrix), counts, and the per-organ
// relation-embedding aggregate relagg[o,c] = sum_{e: org(e)==o} rel_emb[rel(e),c]
__global__ __launch_bounds__(768) void kgt_edges(
    const int* __restrict__ elab, const int* __restrict__ eorg,
    const int* __restrict__ erel, const float* __restrict__ rel_emb,
    float* __restrict__ S, float* __restrict__ counts,
    float* __restrict__ relagg) {
  __shared__ float acc[ODIM * CDIM];  // 48 KB
  int tid = threadIdx.x;
  for (int i = tid; i < ODIM * CDIM; i += 768) acc[i] = 0.0f;
  __syncthreads();
  if (tid < EDIM) {
    int o = eorg[tid];
    atomicAdd(&S[o * LDIM + elab[tid]], 1.0f);
    atomicAdd(&counts[o], 1.0f);
  }
  // each thread exclusively owns channel `tid` of every organ -> no atomics
  for (int e = 0; e < EDIM; ++e) {
    int o = eorg[e];
    int r = erel[e];
    acc[o * CDIM + tid] += rel_emb[(size_t)r * CDIM + tid];
  }
  for (int o = 0; o < ODIM; ++o)
    relagg[o * CDIM + tid] = acc[o * CDIM + tid];
}

// cbias[l,f] = concept[l,:] . W_lab_w[f, F: ] + b_lab[f]
__global__ __launch_bounds__(256) void kgt_cbias(
    const float* __restrict__ concept, const float* __restrict__ W_lab_w,
    const float* __restrict__ W_lab_b, float* __restrict__ cbias) {
  int l = blockIdx.x, f = threadIdx.x;
  float s = W_lab_b[f];
  const float* cr = concept + (size_t)l * CDIM;
  const float* wr = W_lab_w + (size_t)f * (FDIM + CDIM) + FDIM;
  for (int c = 0; c < CDIM; ++c) s += cr[c] * wr[c];
  cbias[l * FDIM + f] = s;
}

// Kmat[o,f] = sum_l S[o,l]*cbias[l,f] + sum_c relagg[o,c]*D_w[f,c]
__global__ __launch_bounds__(256) void kgt_kmat(
    const float* __restrict__ S, const float* __restrict__ cbias,
    const float* __restrict__ relagg, const float* __restrict__ D_w,
    float* __restrict__ Kmat) {
  int o = blockIdx.x, f = threadIdx.x;
  float s = 0.0f;
  for (int l = 0; l < LDIM; ++l) s += S[o * LDIM + l] * cbias[l * FDIM + f];
  const float* ra = relagg + o * CDIM;
  const float* dr = D_w + (size_t)f * CDIM;
  for (int c = 0; c < CDIM; ++c) s += ra[c] * dr[c];
  Kmat[o * FDIM + f] = s;
}

// Kp[o,g] = sum_f W_org[g,f] * Kmat[o,f]
__global__ __launch_bounds__(256) void kgt_kp(const float* __restrict__ W_org,
                                              const float* __restrict__ Kmat,
                                              float* __restrict__ Kp) {
  int o = blockIdx.x, g = threadIdx.x;
  float s = 0.0f;
  const float* wr = W_org + (size_t)g * FDIM;
  const float* kr = Kmat + o * FDIM;
  for (int f = 0; f < FDIM; ++f) s += wr[f] * kr[f];
  Kp[o * GF + g] = s;
}

// M2T[f',g] = sum_f W_org[g,f] * W_lab_w[f, f']   (row-major f' x g for
// coalesced WMMA B-operand loads in the main kernel)
__global__ __launch_bounds__(256) void kgt_m2t(const float* __restrict__ W_org,
                                               const float* __restrict__ W_lab_w,
                                               float* __restrict__ M2T) {
  int fp = blockIdx.x, g = threadIdx.x;
  float s = 0.0f;
  for (int f = 0; f < FDIM; ++f)
    s += W_org[(size_t)g * FDIM + f] * W_lab_w[(size_t)f * (FDIM + CDIM) + fp];
  M2T[fp * GF + g] = s;
}

// ---------------------------------------------------------------------------
// main fused kernel: one workgroup per (b,t); 8 waves x 32 lanes.
// stage B: Y = S @ X      (16x64 . 64x256)   -> Yt in LDS (transposed, padded)
// stage C: out = M2T^T @ Y + epilogue        (16x256 . 256x256 per bt)
// ---------------------------------------------------------------------------
constexpr int YSTRIDE = 20;  // padded row stride (floats) to avoid LDS bank conflicts

__global__ __launch_bounds__(256) void kgt_main(
    const float* __restrict__ lab,        // (B,T,L,F)
    const unsigned char* __restrict__ tmask,  // (B,T) bool
    const float* __restrict__ Sg,         // (O,L)
    const float* __restrict__ counts,     // (O)
    const float* __restrict__ Kp,         // (O,GF)
    const float* __restrict__ M2T,        // (F,GF)
    const float* __restrict__ W_org_b,    // (GF)
    float* __restrict__ out_states,       // (B,T,O,GF)
    float* __restrict__ out_mask) {       // (B,T,O) as 0/1 float
  __shared__ float St[LDIM * ODIM];         // S transposed: St[k*16+o], 4 KB
  __shared__ float Yt[FDIM * YSTRIDE];      // Y transposed: Yt[f*20+o], 20 KB

  const int bt   = blockIdx.x;
  const int tid  = threadIdx.x;
  const int wave = tid >> 5;
  const int lane = tid & 31;
  const int hi   = lane >> 4;   // 0: lanes 0-15, 1: lanes 16-31
  const int ln   = lane & 15;

  for (int i = tid; i < LDIM * ODIM; i += 256) {
    int k = i >> 4, o = i & 15;
    St[i] = Sg[o * LDIM + k];   // transpose on load
  }
  __syncthreads();

  const float* X = lab + (size_t)bt * LDIM * FDIM;  // X[l][f], row-major
  const int n0 = wave * 32;  // this wave's 32-wide column slice

  // ---- stage B: Y[o, n0..n0+31] = sum_l S[o,l] * X[l, n] -------------------
  v8f acc0 = {0.f, 0.f, 0.f, 0.f, 0.f, 0.f, 0.f, 0.f};
  v8f acc1 = acc0;
  for (int k4 = 0; k4 < LDIM; k4 += 4) {
    const int kr = k4 + 2 * hi;  // this half-wave's K rows: kr, kr+1
    // A (16x4 of S): lane ln holds rows M=ln; VGPR v holds K=kr+v
    v2f A = {St[(kr + 0) * ODIM + ln], St[(kr + 1) * ODIM + ln]};
    // B (4x16 of X): lane column n, VGPR v holds K=kr+v
    const float* x0 = X + (size_t)(kr + 0) * FDIM;
    const float* x1 = X + (size_t)(kr + 1) * FDIM;
    v2f B0 = {x0[n0 + ln], x1[n0 + ln]};
    v2f B1 = {x0[n0 + 16 + ln], x1[n0 + 16 + ln]};
    acc0 = wmma_f32(A, B0, acc0);
    acc1 = wmma_f32(A, B1, acc1);
  }
  // C layout: VGPR v -> row o = v + 8*hi, col = ln within tile.
  // store transposed into Yt (8 consecutive o per lane -> two float4 stores)
  {
    float4* p0 = (float4*)&Yt[(n0 + ln) * YSTRIDE + 8 * hi];
    float4* p1 = (float4*)&Yt[(n0 + 16 + ln) * YSTRIDE + 8 * hi];
    p0[0] = make_float4(acc0[0], acc0[1], acc0[2], acc0[3]);
    p0[1] = make_float4(acc0[4], acc0[5], acc0[6], acc0[7]);
    p1[0] = make_float4(acc1[0], acc1[1], acc1[2], acc1[3]);
    p1[1] = make_float4(acc1[4], acc1[5], acc1[6], acc1[7]);
  }
  __syncthreads();

  // ---- stage C: out[o, g0..g0+31] = sum_f Y[o,f] * M2T[f,g] ----------------
  const int g0 = n0;
  v8f o0 = {0.f, 0.f, 0.f, 0.f, 0.f, 0.f, 0.f, 0.f};
  v8f o1 = o0;
  for (int k4 = 0; k4 < FDIM; k4 += 4) {
    const int kr = k4 + 2 * hi;
    v2f A = {Yt[(kr + 0) * YSTRIDE + ln], Yt[(kr + 1) * YSTRIDE + ln]};
    const float* m0 = M2T + (size_t)(kr + 0) * GF;
    const float* m1 = M2T + (size_t)(kr + 1) * GF;
    v2f B0 = {m0[g0 + ln], m1[g0 + ln]};
    v2f B1 = {m0[g0 + 16 + ln], m1[g0 + 16 + ln]};
    o0 = wmma_f32(A, B0, o0);
    o1 = wmma_f32(A, B1, o1);
  }

  // ---- epilogue: scale by mask/count, add Kp and bias ----------------------
  const float m = (tmask[bt] != 0) ? 1.0f : 0.0f;
  float* outbt = out_states + (size_t)bt * ODIM * GF;
#pragma unroll
  for (int v = 0; v < 8; ++v) {
    const int org = v + 8 * hi;
    const float scale = m / fmaxf(counts[org], 1.0f);
    const int ga = g0 + ln;
    const int gb = g0 + 16 + ln;
    outbt[org * GF + ga] = scale * (o0[v] + Kp[org * GF + ga]) + W_org_b[ga];
    outbt[org * GF + gb] = scale * (o1[v] + Kp[org * GF + gb]) + W_org_b[gb];
  }
  if (tid < ODIM)
    out_mask[(size_t)bt * ODIM + tid] =
        (m != 0.0f && counts[tid] > 0.0f) ? 1.0f : 0.0f;
}

// ---------------------------------------------------------------------------
extern "C" void kernel_launch(void* const* d_in, const int* in_sizes, int n_in,
                              void* d_out, int out_size, void* d_ws,
                              size_t ws_size, hipStream_t stream) {
  const float* lab          = (const float*)d_in[0];
  const unsigned char* msk  = (const unsigned char*)d_in[1];  // jnp bool = 1B
  const int* elab           = (const int*)d_in[2];
  const int* eorg           = (const int*)d_in[3];
  const int* erel           = (const int*)d_in[4];
  // d_in[5] = num_organs (fixed at 16 for these shapes)
  const float* concept      = (const float*)d_in[6];
  const float* rel_emb      = (const float*)d_in[7];
  const float* W_lab_w      = (const float*)d_in[8];
  const float* W_lab_b      = (const float*)d_in[9];
  const float* D_w          = (const float*)d_in[10];
  const float* W_org_w      = (const float*)d_in[11];
  const float* W_org_b      = (const float*)d_in[12];

  float* ws     = (float*)d_ws;
  float* S      = ws + OFF_S;
  float* counts = ws + OFF_CNT;
  float* relagg = ws + OFF_RELAGG;
  float* cbias  = ws + OFF_CBIAS;
  float* Kmat   = ws + OFF_KMAT;
  float* Kp     = ws + OFF_KP;
  float* M2T    = ws + OFF_M2T;

  float* out_states = (float*)d_out;
  float* out_mask   = out_states + (size_t)BT * ODIM * GF;

  // setup (tiny, serialized on stream)
  kgt_zero<<<(OFF_RELAGG + 255) / 256, 256, 0, stream>>>(ws, OFF_RELAGG);
  kgt_edges<<<1, 768, 0, stream>>>(elab, eorg, erel, rel_emb, S, counts, relagg);
  kgt_cbias<<<LDIM, 256, 0, stream>>>(concept, W_lab_w, W_lab_b, cbias);
  kgt_kmat<<<ODIM, 256, 0, stream>>>(S, cbias, relagg, D_w, Kmat);
  kgt_kp<<<ODIM, 256, 0, stream>>>(W_org_w, Kmat, Kp);
  kgt_m2t<<<FDIM, 256, 0, stream>>>(W_org_w, W_lab_w, M2T);

  // main fused kernel: one workgroup per (b,t)
  kgt_main<<<BT, 256, 0, stream>>>(lab, msk, S, counts, Kp, M2T, W_org_b,
                                   out_states, out_mask);
}